// CrossLocalAttentionLayer_33895881900516
// MI455X (gfx1250) — compile-verified
//
#include <hip/hip_runtime.h>
#include <hip/hip_bf16.h>
#include <math.h>
#include <stdint.h>

// Problem constants (from reference)
#define Bq  4
#define Nn  16384
#define EGe 131072
#define Hh  4
#define DMm 128          // model dim; DK = DV = 32, H*DK = 128

typedef __attribute__((ext_vector_type(2))) float v2f;
typedef __attribute__((ext_vector_type(4))) float v4f;
typedef __attribute__((ext_vector_type(8))) float v8f;

static constexpr int TPB        = 256;  // 8 waves (wave32)
static constexpr int ROW_TILES  = 8;    // 16-row tiles per block in GEMM kernels
static constexpr int EDGE_TILES = 8;    // 16-edge tiles per block in edge kernel

// ---------------------------------------------------------------------------
// Preload per-lane B-matrix fragments for a 128(K) x 16(N) stripe of W.
// V_WMMA_F32_16X16X4_F32 B layout (mirror of A, ISA 7.12.2):
//   lanes 0-15  : col = lane,    VGPR0 = K=k0+0, VGPR1 = K=k0+1
//   lanes 16-31 : col = lane-16, VGPR0 = K=k0+2, VGPR1 = K=k0+3
// ---------------------------------------------------------------------------
__device__ __forceinline__ void load_bfrag(const float* __restrict__ W,
                                           int col, int khalf, v2f* bf) {
#pragma unroll
  for (int k = 0; k < 32; ++k) {
    int kk = 4 * k + khalf;
    v2f t;
    t.x = W[kk * DMm + col];
    t.y = W[(kk + 1) * DMm + col];
    bf[k] = t;
  }
}

// 16x16 f32 output tile = (16x128 LDS A-tile) x (preloaded 128x16 B fragments)
__device__ __forceinline__ v8f mma16x128(const float* __restrict__ sA,
                                         const v2f* __restrict__ bf, int lane) {
  v8f acc = {};
  const int r     = lane & 15;
  const int khalf = (lane >> 4) << 1;
#pragma unroll
  for (int k = 0; k < 32; ++k) {
    const int kk = 4 * k + khalf;
    v2f a;
    a.x = sA[r * DMm + kk];
    a.y = sA[r * DMm + kk + 1];
    acc = __builtin_amdgcn_wmma_f32_16x16x4_f32(false, a, false, bf[k],
                                                (short)0, acc, false, false);
  }
  return acc;
}

// ---------------------------------------------------------------------------
// Async global->LDS copy of one 16x128 f32 tile (8KB), 16B per lane per op.
// CDNA5 GLOBAL_LOAD_ASYNC_TO_LDS_B128 (tracked by ASYNCcnt); the immediate
// offset is added to BOTH the global and LDS address per the ISA formula,
// so one address VGPR pair covers both 4KB halves.
// ---------------------------------------------------------------------------
__device__ __forceinline__ void stage_tile_async(const float* __restrict__ gbase,
                                                 uint32_t lds_off, int tid) {
  uint64_t g = (uint64_t)(uintptr_t)(gbase + tid * 4);
  uint32_t l = lds_off + tid * 16;
  asm volatile("global_load_async_to_lds_b128 %0, %1, off"
               :: "v"(l), "v"(g) : "memory");
  asm volatile("global_load_async_to_lds_b128 %0, %1, off offset:4096"
               :: "v"(l), "v"(g) : "memory");
}

__device__ __forceinline__ void wait_async_le2() {
  asm volatile("s_wait_asynccnt 0x2" ::: "memory");
}
__device__ __forceinline__ void wait_async_0() {
  asm volatile("s_wait_asynccnt 0x0" ::: "memory");
}

// ---------------------------------------------------------------------------
__global__ void zero_kernel(float* __restrict__ p, long n) {
  long i = (long)blockIdx.x * blockDim.x + threadIdx.x;
  if (i < n) p[i] = 0.0f;
}

// Y(nrows x 128) = X(nrows x 128) @ W(128 x 128), f32 WMMA,
// double-buffered async A-tile staging. nrows must divide 16*ROW_TILES*grid.
__global__ __launch_bounds__(TPB)
void proj_kernel(const float* __restrict__ X, const float* __restrict__ W,
                 float* __restrict__ Y) {
  __shared__ __align__(16) float sA[2][16 * DMm];
  const int tid   = threadIdx.x;
  const int lane  = tid & 31;
  const int wave  = tid >> 5;
  const int col   = wave * 16 + (lane & 15);
  const int khalf = (lane >> 4) << 1;
  const uint32_t ldsA[2] = {(uint32_t)(uintptr_t)&sA[0][0],
                            (uint32_t)(uintptr_t)&sA[1][0]};

  const long base = (long)blockIdx.x * ROW_TILES * 16;
  stage_tile_async(X + base * DMm, ldsA[0], tid);   // prefetch tile 0

  v2f bf[32];
  load_bfrag(W, col, khalf, bf);                    // overlaps with async copy

  for (int t = 0; t < ROW_TILES; ++t) {
    const long row0 = base + (long)t * 16;
    if (t + 1 < ROW_TILES) {
      stage_tile_async(X + (row0 + 16) * DMm, ldsA[(t + 1) & 1], tid);
      wait_async_le2();   // async loads complete in order -> tile t is in LDS
    } else {
      wait_async_0();
    }
    __syncthreads();

    v8f acc = mma16x128(sA[t & 1], bf, lane);

#pragma unroll
    for (int v = 0; v < 8; ++v) {
      const int m = v + ((lane >> 4) << 3);   // C/D: VGPR v -> row v or v+8
      Y[(row0 + m) * DMm + col] = acc[v];
    }
    __syncthreads();
  }
}

// Fused: E = edge_features @ W_E (WMMA, kept in LDS) + gather/score/scatter.
// Edge-feature tiles double-buffered via async-to-LDS; indices staged once.
__global__ __launch_bounds__(TPB)
void edge_kernel(const int* __restrict__ eidx, const float* __restrict__ efeat,
                 const float* __restrict__ WE,
                 const float* __restrict__ Qw, const float* __restrict__ Kw,
                 const float* __restrict__ Vw,
                 float* __restrict__ num, float* __restrict__ den,
                 float* __restrict__ attn_out) {
  __shared__ __align__(16) float sF[2][16 * DMm];  // raw edge-feature tiles
  __shared__ __align__(16) float sE[16 * DMm];     // projected E tile
  __shared__ int sSrc[16];
  __shared__ int sTgt[16];
  const int tid   = threadIdx.x;
  const int lane  = tid & 31;
  const int wave  = tid >> 5;
  const int col   = wave * 16 + (lane & 15);
  const int khalf = (lane >> 4) << 1;
  const int b     = blockIdx.y;
  const float inv_sqrt_dk = 0.17677669529663687f;  // 1/sqrt(32)
  const uint32_t ldsF[2] = {(uint32_t)(uintptr_t)&sF[0][0],
                            (uint32_t)(uintptr_t)&sF[1][0]};

  const float* fbase = efeat + (long)b * EGe * DMm;
  const long   e00   = (long)blockIdx.x * (16 * EDGE_TILES);
  stage_tile_async(fbase + e00 * DMm, ldsF[0], tid);   // prefetch tile 0

  v2f bf[32];
  load_bfrag(WE, col, khalf, bf);

  for (int t = 0; t < EDGE_TILES; ++t) {
    const long e0 = e00 + (long)t * 16;

    if (t + 1 < EDGE_TILES) {
      stage_tile_async(fbase + (e0 + 16) * DMm, ldsF[(t + 1) & 1], tid);
      wait_async_le2();
    } else {
      wait_async_0();
    }
    // stage the 16 src / 16 tgt indices once per tile (wave 0)
    if (tid < 16)       sSrc[tid]      = eidx[(long)b * 2 * EGe + e0 + tid];
    else if (tid < 32)  sTgt[tid - 16] = eidx[((long)b * 2 + 1) * EGe + e0 + tid - 16];
    __syncthreads();

    v8f acc = mma16x128(sF[t & 1], bf, lane);
#pragma unroll
    for (int v = 0; v < 8; ++v) {
      const int m = v + ((lane >> 4) << 3);
      sE[m * DMm + col] = acc[v];
    }
    __syncthreads();

    // 16 threads per edge: j = h*4 + quarter; each quarter handles 8 dims
    const int el = tid >> 4;            // 0..15 local edge
    const int j  = tid & 15;
    const int h  = j >> 2;
    const int d0 = h * 32 + (j & 3) * 8;
    const long eg = e0 + el;
    const int srcn = sSrc[el];
    const int tgtn = sTgt[el];

    const float* qrow = Qw + ((long)b * Nn + srcn) * DMm;
    const float* krow = Kw + ((long)b * Nn + tgtn) * DMm;
    const float* vrow = Vw + ((long)b * Nn + tgtn) * DMm;

    float part = 0.0f;
#pragma unroll
    for (int i = 0; i < 8; ++i)
      part += qrow[d0 + i] * krow[d0 + i] * sE[el * DMm + d0 + i];
    part += __shfl_xor(part, 1, 32);
    part += __shfl_xor(part, 2, 32);

    float sc = part * inv_sqrt_dk;
    sc = fminf(5.0f, fmaxf(-5.0f, sc));
    const float a = expf(sc);

#pragma unroll
    for (int i = 0; i < 8; ++i)
      atomicAdd(&num[((long)b * Nn + srcn) * DMm + d0 + i], a * vrow[d0 + i]);
    if ((j & 3) == 0) {
      atomicAdd(&den[((long)b * Nn + srcn) * Hh + h], a);
      if (b == Bq - 1) attn_out[(long)h * EGe + eg] = a;  // attn[-1:] output
    }
    __syncthreads();
  }
}

// out = LayerNorm( (num/(den+1e-8)) @ W_fc + residual )
__global__ __launch_bounds__(TPB)
void finalize_kernel(const float* __restrict__ num, const float* __restrict__ den,
                     const float* __restrict__ Wfc, const float* __restrict__ resid,
                     const float* __restrict__ gamma, const float* __restrict__ beta,
                     float* __restrict__ outp) {
  __shared__ __align__(16) float sA[16 * DMm];
  __shared__ __align__(16) float sO[16 * DMm];
  const int tid   = threadIdx.x;
  const int lane  = tid & 31;
  const int wave  = tid >> 5;
  const int col   = wave * 16 + (lane & 15);
  const int khalf = (lane >> 4) << 1;

  v2f bf[32];
  load_bfrag(Wfc, col, khalf, bf);

  for (int t = 0; t < ROW_TILES; ++t) {
    const long row0 = ((long)blockIdx.x * ROW_TILES + t) * 16;

    // build normalized A tile: num / (den + eps); 8 contiguous elems / thread
    {
      const int flat = tid * 8;
      const int r = flat >> 7;
      const int c = flat & 127;
      const float dv   = den[(long)(row0 + r) * Hh + (c >> 5)] + 1e-8f;
      const float invd = 1.0f / dv;
      const float* nrow = num + (long)(row0 + r) * DMm + c;
#pragma unroll
      for (int i = 0; i < 8; ++i) sA[flat + i] = nrow[i] * invd;
    }
    __syncthreads();

    v8f acc = mma16x128(sA, bf, lane);
#pragma unroll
    for (int v = 0; v < 8; ++v) {
      const int m = v + ((lane >> 4) << 3);
      sO[m * DMm + col] = acc[v] + resid[(row0 + m) * DMm + col];
    }
    __syncthreads();

    // LayerNorm over 128 cols; 16 threads per row, shuffle tree reduction
    {
      const int r = tid >> 4;
      const int j = tid & 15;
      float s = 0.0f;
#pragma unroll
      for (int i = 0; i < 8; ++i) s += sO[r * DMm + j + 16 * i];
#pragma unroll
      for (int m = 1; m < 16; m <<= 1) s += __shfl_xor(s, m, 32);
      const float mu = s * (1.0f / 128.0f);

      float s2 = 0.0f;
#pragma unroll
      for (int i = 0; i < 8; ++i) {
        const float d = sO[r * DMm + j + 16 * i] - mu;
        s2 += d * d;
      }
#pragma unroll
      for (int m = 1; m < 16; m <<= 1) s2 += __shfl_xor(s2, m, 32);
      const float rstd = rsqrtf(s2 * (1.0f / 128.0f) + 1e-5f);

#pragma unroll
      for (int i = 0; i < 8; ++i) {
        const int c = j + 16 * i;
        const float x = (sO[r * DMm + c] - mu) * rstd;
        outp[(row0 + r) * DMm + c] = x * gamma[c] + beta[c];
      }
    }
    __syncthreads();
  }
}

// ---------------------------------------------------------------------------
extern "C" void kernel_launch(void* const* d_in, const int* in_sizes, int n_in,
                              void* d_out, int out_size, void* d_ws, size_t ws_size,
                              hipStream_t stream) {
  (void)in_sizes; (void)n_in; (void)out_size; (void)ws_size;

  const int*   eidx  = (const int*)  d_in[0];
  const float* efeat = (const float*)d_in[1];
  const float* inQ   = (const float*)d_in[2];
  const float* inK   = (const float*)d_in[3];
  const float* inV   = (const float*)d_in[4];
  const float* WQ    = (const float*)d_in[5];
  const float* WK    = (const float*)d_in[6];
  const float* WV    = (const float*)d_in[7];
  const float* WE    = (const float*)d_in[8];
  const float* Wfc   = (const float*)d_in[9];
  const float* gamma = (const float*)d_in[10];
  const float* beta  = (const float*)d_in[11];

  float* out      = (float*)d_out;
  float* attn_out = out + (size_t)Bq * Nn * DMm;   // outputs concatenated flat

  // workspace layout (floats): Q | K | V | num | den
  const size_t nodeElems = (size_t)Bq * Nn * DMm;  // 8.4M floats each
  float* Qw   = (float*)d_ws;
  float* Kw   = Qw + nodeElems;
  float* Vw   = Kw + nodeElems;
  float* numw = Vw + nodeElems;
  float* denw = numw + nodeElems;                  // (B,N,H), contiguous after num

  const long accN = (long)nodeElems + (long)Bq * Nn * Hh;
  zero_kernel<<<(int)((accN + 255) / 256), 256, 0, stream>>>(numw, accN);

  const long nodeRows = (long)Bq * Nn;                       // 65536
  const int  projGrid = (int)(nodeRows / (16 * ROW_TILES));  // 512 (exact)

  proj_kernel<<<projGrid, TPB, 0, stream>>>(inQ, WQ, Qw);
  proj_kernel<<<projGrid, TPB, 0, stream>>>(inK, WK, Kw);
  proj_kernel<<<projGrid, TPB, 0, stream>>>(inV, WV, Vw);

  dim3 egrid(EGe / (16 * EDGE_TILES), Bq);                   // (1024, 4) exact
  edge_kernel<<<egrid, TPB, 0, stream>>>(eidx, efeat, WE, Qw, Kw, Vw,
                                         numw, denw, attn_out);

  finalize_kernel<<<projGrid, TPB, 0, stream>>>(numw, denw, Wfc, inQ,
                                                gamma, beta, out);
}